// EncoderLSTM_52613349376072
// MI455X (gfx1250) — compile-verified
//
#include <hip/hip_runtime.h>
#include <math.h>

// CDNA5 / gfx1250: wave32, WMMA 16x16x4 f32 + async global->LDS staging.
typedef __attribute__((ext_vector_type(2))) float v2f;
typedef __attribute__((ext_vector_type(4))) float v4f;
typedef __attribute__((ext_vector_type(8))) float v8f;
typedef __attribute__((ext_vector_type(4))) int   v4i;

#define HSZ   1024
#define BATCH 64
#define NENDO 48
#define FDIM  64
#define LOUT  64
#define LIN   256

#define KC   64            // K-chunk staged per double-buffer step
#define LDR  (KC + 4)      // padded LDS row stride (floats): 68 -> 4-bank rotation per row

#define AS1 __attribute__((address_space(1)))
#define AS3 __attribute__((address_space(3)))

#if __has_builtin(__builtin_amdgcn_global_load_async_to_lds_b128)
#define HAVE_ASYNC 1
// Signature (from clang diagnostics): (AS1 v4i*, AS3 v4i*, imm offset, imm cpol).
// Cast generic pointers -> v4i* -> address-space-qualified (emits addrspacecast).
#define ASYNC_CP(gp, lp)                                                        \
    __builtin_amdgcn_global_load_async_to_lds_b128(                             \
        (AS1 v4i*)(v4i*)(gp), (AS3 v4i*)(v4i*)(lp), 0, 0)
#if __has_builtin(__builtin_amdgcn_s_wait_asynccnt)
#define ASYNC_WAIT(n) __builtin_amdgcn_s_wait_asynccnt(n)
#else
#define ASYNC_WAIT(n) asm volatile("s_wait_asynccnt %0" ::"n"(n) : "memory")
#endif
#else
#define HAVE_ASYNC 0
// Fallback: synchronous stage through VGPRs (global_load_b128 + ds_store_b128).
#define ASYNC_CP(gp, lp) (*(v4f*)(lp) = *(const v4f*)(gp))
#define ASYNC_WAIT(n) do { } while (0)
#endif

__device__ __forceinline__ float sigmoidf_(float x) { return 1.0f / (1.0f + __expf(-x)); }

// ---------------------------------------------------------------------------
// One LSTM cell step:
//   gates = x @ Wih^T + h_in @ Whh^T + bih + bhh   (gates order i,f,g,o)
//   c = sig(f)*c + sig(i)*tanh(g); h_out = sig(o)*tanh(c)
// grid: 64 blocks (16-wide j-tiles over H), block = 128 threads = 4 waves.
// Each chunk of K (KC=64) is staged into LDS double-buffered via async DMA:
//   As: all 64 batch rows of x/h, Bs: 64 weight rows (4 gates x 16 cols).
// ---------------------------------------------------------------------------
__global__ __launch_bounds__(128) void lstm_cell_step(
    const float* __restrict__ x, int x_stride, int xK,
    const float* __restrict__ Wih,                 // (4H, xK) row-major
    const float* __restrict__ h_in,                // (B, H)
    const float* __restrict__ Whh,                 // (4H, H) row-major
    const float* __restrict__ bih,
    const float* __restrict__ bhh,
    float* __restrict__ c_state,                   // (B, H), in-place
    float* __restrict__ h_out)                     // (B, H)
{
    __shared__ float As[2][BATCH][LDR];
    __shared__ float Bs[2][64][LDR];

    const int tid  = threadIdx.x;
    const int lane = tid & 31;
    const int wave = tid >> 5;            // batch M-tile 0..3
    const int j0   = blockIdx.x * 16;     // column tile base within H
    const int mb   = wave * 16;
    const int ln   = lane & 15;
    const int koff = (lane >> 4) * 2;     // ISA 16x16x4 f32 A/B per-lane K sub-offset
    const int m    = mb + ln;             // A-matrix row this lane feeds

    const int nx = xK / KC;               // chunks from the x@Wih part (1 or 16)
    const int NC = nx + HSZ / KC;         // total K chunks

    // stage one K-chunk (c) into buffer bf: 16 x b128 per thread
    auto issue = [&](int c, int bf) {
        const float* asrc;
        const float* W;
        int astride, Kd, kb;
        if (c < nx) { asrc = x;    astride = x_stride; W = Wih; Kd = xK;  kb = c * KC; }
        else        { asrc = h_in; astride = HSZ;      W = Whh; Kd = HSZ; kb = (c - nx) * KC; }
#pragma unroll
        for (int s = 0; s < 8; ++s) {
            const int q    = tid + 128 * s;      // 0..1023
            const int row  = q >> 4;             // 0..63
            const int slot = q & 15;             // 16B slot within the 64-float row
            const float* ga = asrc + (size_t)row * astride + kb + slot * 4;
            ASYNC_CP(ga, &As[bf][row][slot * 4]);
            const int wrow = (row >> 4) * HSZ + j0 + (row & 15);
            const float* gb = W + (size_t)wrow * Kd + kb + slot * 4;
            ASYNC_CP(gb, &Bs[bf][row][slot * 4]);
        }
    };

    v8f acc[4] = {v8f{}, v8f{}, v8f{}, v8f{}};

    issue(0, 0);
    for (int c = 0; c < NC; ++c) {
        const int bf = c & 1;
        if (c + 1 < NC) { issue(c + 1, bf ^ 1); ASYNC_WAIT(16); }
        else            { ASYNC_WAIT(0); }
        __syncthreads();                   // chunk c fully in LDS for all waves

#pragma unroll
        for (int k0 = 0; k0 < KC; k0 += 4) {
            v2f a = *(const v2f*)&As[bf][m][k0 + koff];
#pragma unroll
            for (int g = 0; g < 4; ++g) {
                v2f b = *(const v2f*)&Bs[bf][g * 16 + ln][k0 + koff];
                acc[g] = __builtin_amdgcn_wmma_f32_16x16x4_f32(false, a, false, b,
                                                               (short)0, acc[g], false, false);
            }
        }
        __syncthreads();                   // protect buffer bf before it is re-filled
    }

    // ---- elementwise LSTM update (each wave owns its 16x16 (b,j) tile) ----
    const int j = j0 + ln;                 // C/D: N = lane&15
    const float bi = bih[j]           + bhh[j];
    const float bf_ = bih[HSZ + j]    + bhh[HSZ + j];
    const float bg = bih[2 * HSZ + j] + bhh[2 * HSZ + j];
    const float bo = bih[3 * HSZ + j] + bhh[3 * HSZ + j];
    const int mhi = 8 * (lane >> 4);       // C/D: M = r + 8*(lane>=16)

#pragma unroll
    for (int r = 0; r < 8; ++r) {
        const int bm = mb + r + mhi;
        const float iv = sigmoidf_(acc[0][r] + bi);
        const float fv = sigmoidf_(acc[1][r] + bf_);
        const float gv = tanhf(acc[2][r] + bg);
        const float ov = sigmoidf_(acc[3][r] + bo);
        const size_t idx = (size_t)bm * HSZ + j;
        const float cn = fv * c_state[idx] + iv * gv;
        c_state[idx] = cn;
        h_out[idx]   = ov * tanhf(cn);
    }
}

// ---------------------------------------------------------------------------
// Decoder FC + feedback: pred(B x 48) = h1 @ fc_W^T + fc_b
//   out[:,t,:] = pred ; xbuf = concat(pred, trg[:, t, 48:64])
// 1 block x 128 threads = 4 waves; wave = M-tile, each computes 3 N-tiles.
// Same async double-buffered LDS staging over K=1024 (16 chunks).
// ---------------------------------------------------------------------------
__global__ __launch_bounds__(128) void dec_fc_step(
    const float* __restrict__ h1,      // (B, H)
    const float* __restrict__ fcW,     // (48, H)
    const float* __restrict__ fcb,     // (48)
    const float* __restrict__ trg,     // (B, LOUT, F)
    float* __restrict__ out,           // (B, LOUT, 48)
    float* __restrict__ xbuf,          // (B, F)
    int t)
{
    __shared__ float As[2][BATCH][LDR];
    __shared__ float Bs[2][NENDO][LDR];

    const int tid  = threadIdx.x;
    const int lane = tid & 31;
    const int wave = tid >> 5;            // M-tile 0..3
    const int mb   = wave * 16;
    const int ln   = lane & 15;
    const int koff = (lane >> 4) * 2;
    const int m    = mb + ln;
    const int NC   = HSZ / KC;            // 16 chunks

    auto issue = [&](int c, int bf) {
        const int kb = c * KC;
        // A: 64 rows x 16 slots = 1024 b128;  B: 48 rows x 16 slots = 768 b128
#pragma unroll
        for (int s = 0; s < 14; ++s) {
            const int q = tid + 128 * s;          // 0..1791
            if (q < 1024) {
                const int row = q >> 4, slot = q & 15;
                ASYNC_CP(h1 + (size_t)row * HSZ + kb + slot * 4, &As[bf][row][slot * 4]);
            } else {
                const int qb = q - 1024;
                const int row = qb >> 4, slot = qb & 15;
                ASYNC_CP(fcW + (size_t)row * HSZ + kb + slot * 4, &Bs[bf][row][slot * 4]);
            }
        }
    };

    v8f acc[3] = {v8f{}, v8f{}, v8f{}};

    issue(0, 0);
    for (int c = 0; c < NC; ++c) {
        const int bf = c & 1;
        if (c + 1 < NC) { issue(c + 1, bf ^ 1); ASYNC_WAIT(14); }
        else            { ASYNC_WAIT(0); }
        __syncthreads();

#pragma unroll
        for (int k0 = 0; k0 < KC; k0 += 4) {
            v2f a = *(const v2f*)&As[bf][m][k0 + koff];
#pragma unroll
            for (int nt = 0; nt < 3; ++nt) {
                v2f b = *(const v2f*)&Bs[bf][nt * 16 + ln][k0 + koff];
                acc[nt] = __builtin_amdgcn_wmma_f32_16x16x4_f32(false, a, false, b,
                                                                (short)0, acc[nt], false, false);
            }
        }
        __syncthreads();
    }

    const int mhi = 8 * (lane >> 4);
#pragma unroll
    for (int nt = 0; nt < 3; ++nt) {
        const int n = nt * 16 + ln;
        const float bb = fcb[n];
#pragma unroll
        for (int r = 0; r < 8; ++r) {
            const int bm = mb + r + mhi;
            const float v = acc[nt][r] + bb;
            out[((size_t)bm * LOUT + t) * NENDO + n] = v;
            xbuf[bm * FDIM + n] = v;
        }
    }

    // exogenous copy: xbuf[b, 48+e] = trg[b, t, 48+e]
    for (int idx = tid; idx < BATCH * (FDIM - NENDO); idx += 128) {
        const int b = idx >> 4;
        const int e = idx & 15;
        xbuf[b * FDIM + NENDO + e] = trg[((size_t)b * LOUT + t) * FDIM + NENDO + e];
    }
}

__global__ void zero_ws(float* __restrict__ p, int n) {
    int i = blockIdx.x * blockDim.x + threadIdx.x;
    if (i < n) p[i] = 0.0f;
}

// ---------------------------------------------------------------------------
// Host orchestration. Inputs (setup_inputs order):
// 0 src, 1 trg, 2..5 enc0 (Wih,Whh,bih,bhh), 6..9 enc1, 10..13 dec0,
// 14..17 dec1, 18 fc_W, 19 fc_b
// ---------------------------------------------------------------------------
extern "C" void kernel_launch(void* const* d_in, const int* in_sizes, int n_in,
                              void* d_out, int out_size, void* d_ws, size_t ws_size,
                              hipStream_t stream) {
    const float* src = (const float*)d_in[0];
    const float* trg = (const float*)d_in[1];
    const float* eW0[4] = {(const float*)d_in[2],  (const float*)d_in[3],
                           (const float*)d_in[4],  (const float*)d_in[5]};
    const float* eW1[4] = {(const float*)d_in[6],  (const float*)d_in[7],
                           (const float*)d_in[8],  (const float*)d_in[9]};
    const float* dW0[4] = {(const float*)d_in[10], (const float*)d_in[11],
                           (const float*)d_in[12], (const float*)d_in[13]};
    const float* dW1[4] = {(const float*)d_in[14], (const float*)d_in[15],
                           (const float*)d_in[16], (const float*)d_in[17]};
    const float* fcW = (const float*)d_in[18];
    const float* fcb = (const float*)d_in[19];
    float* out = (float*)d_out;

    const int ST = BATCH * HSZ;            // 65536 floats per state buffer
    float* ws   = (float*)d_ws;
    float* h0[2] = {ws,          ws + ST};
    float* h1[2] = {ws + 2 * ST, ws + 3 * ST};
    float* c0    = ws + 4 * ST;
    float* c1    = ws + 5 * ST;
    float* xbuf  = ws + 6 * ST;            // (B, F) = 4096 floats

    const int nzero = 6 * ST + BATCH * FDIM;
    zero_ws<<<(nzero + 255) / 256, 256, 0, stream>>>(ws, nzero);

    const dim3 cgrid(HSZ / 16), cblk(128);
    int p = 0;

    // ---- encoder: 256 steps x 2 layers ----
    for (int t = 0; t < LIN; ++t) {
        lstm_cell_step<<<cgrid, cblk, 0, stream>>>(
            src + (size_t)t * FDIM, LIN * FDIM, FDIM,
            eW0[0], h0[p], eW0[1], eW0[2], eW0[3], c0, h0[p ^ 1]);
        lstm_cell_step<<<cgrid, cblk, 0, stream>>>(
            h0[p ^ 1], HSZ, HSZ,
            eW1[0], h1[p], eW1[1], eW1[2], eW1[3], c1, h1[p ^ 1]);
        p ^= 1;
    }

    // ---- decoder: 64 steps x (2 layers + FC feedback) ----
    for (int t = 0; t < LOUT; ++t) {
        const float* x = (t == 0) ? (src + (size_t)(LIN - 1) * FDIM) : xbuf;
        const int xs   = (t == 0) ? (LIN * FDIM) : FDIM;
        lstm_cell_step<<<cgrid, cblk, 0, stream>>>(
            x, xs, FDIM,
            dW0[0], h0[p], dW0[1], dW0[2], dW0[3], c0, h0[p ^ 1]);
        lstm_cell_step<<<cgrid, cblk, 0, stream>>>(
            h0[p ^ 1], HSZ, HSZ,
            dW1[0], h1[p], dW1[1], dW1[2], dW1[3], c1, h1[p ^ 1]);
        dec_fc_step<<<1, 128, 0, stream>>>(h1[p ^ 1], fcW, fcb, trg, out, xbuf, t);
        p ^= 1;
    }
}